// IoU3DLoss_20392504722039
// MI455X (gfx1250) — compile-verified
//
#include <hip/hip_runtime.h>
#include <hip/hip_bf16.h>
#include <math.h>

typedef float v2f __attribute__((ext_vector_type(2)));
typedef float v8f __attribute__((ext_vector_type(8)));

// ---------------------------------------------------------------------------
// Corner generation: matches box2corners() in the reference.
// ---------------------------------------------------------------------------
__device__ __forceinline__ void box_corners(float x, float y, float w, float h,
                                            float a, float cx[4], float cy[4]) {
  const float xs[4] = {0.5f, -0.5f, -0.5f, 0.5f};
  const float ys[4] = {0.5f, 0.5f, -0.5f, -0.5f};
  float c = cosf(a), s = sinf(a);
#pragma unroll
  for (int k = 0; k < 4; ++k) {
    float X = xs[k] * w, Y = ys[k] * h;
    cx[k] = X * c - Y * s + x;
    cy[k] = X * s + Y * c + y;
  }
}

// ---------------------------------------------------------------------------
// Main kernel: one thread per box pair. Computes
//   contrib = (|pred_iou - (2*iou-1)| + (1 - diou)) * mask
// then block-reduces (shfl within wave32, LDS across the 8 waves) into two
// per-block partials: partials[blk] = sum contrib, partials[nbp+blk] = sum mask.
// ---------------------------------------------------------------------------
__global__ void __launch_bounds__(256) iou3d_loss_main_kernel(
    const float* __restrict__ pred_iou,
    const float* __restrict__ pred_boxes,   // (total, 7)
    const float* __restrict__ gt_boxes,     // (total, 8)
    const unsigned char* __restrict__ mask, // (total,) bool
    float* __restrict__ partials,           // [0,nbp): num, [nbp,2*nbp): cnt
    int total, int nbp)
{
  int idx = blockIdx.x * blockDim.x + threadIdx.x;
  float contrib = 0.0f, cntf = 0.0f;

  if (idx < total) {
    const float* pb = pred_boxes + (size_t)idx * 7;
    const float* gb = gt_boxes + (size_t)idx * 8;
    float px = pb[0], py = pb[1], pz = pb[2], pw = pb[3], ph = pb[4], pd = pb[5], pa = pb[6];
    float gx = gb[0], gy = gb[1], gz = gb[2], gw = gb[3], gh = gb[4], gd = gb[5];
    float ga = atan2f(gb[6], gb[7]);

    float c1x[4], c1y[4], c2x[4], c2y[4];
    box_corners(px, py, pw, ph, pa, c1x, c1y);
    box_corners(gx, gy, gw, gh, ga, c2x, c2y);

    // 24 candidate vertices: 4 + 4 corners, 16 edge intersections
    float vx[24], vy[24], ang[24];
    bool vm[24];
#pragma unroll
    for (int k = 0; k < 4; ++k) {
      vx[k] = c1x[k];     vy[k] = c1y[k];
      vx[4 + k] = c2x[k]; vy[4 + k] = c2y[k];
    }

    // --- box_intersection: 4x4 segment intersections ---
#pragma unroll
    for (int i = 0; i < 4; ++i) {
      float x1 = c1x[i], y1 = c1y[i];
      float x2 = c1x[(i + 1) & 3], y2 = c1y[(i + 1) & 3];
#pragma unroll
      for (int j = 0; j < 4; ++j) {
        float x3 = c2x[j], y3 = c2y[j];
        float x4 = c2x[(j + 1) & 3], y4 = c2y[(j + 1) & 3];
        float den  = (x4 - x3) * (y1 - y2) - (x1 - x2) * (y4 - y3);
        float numt = (y4 - y3) * (x1 - x3) - (x4 - x3) * (y1 - y3);
        float numu = (y1 - y2) * (x1 - x3) - (x1 - x2) * (y1 - y3);
        bool dz = (den == 0.0f);
        float t = dz ? -1.0f : (numt / den);
        float u = dz ? -1.0f : (-numu / den);
        bool m = (t > 0.0f) && (t < 1.0f) && (u > 0.0f) && (u < 1.0f);
        int o = 8 + i * 4 + j;
        vx[o] = m ? (x1 + t * (x2 - x1)) : 0.0f;   // pts * mask
        vy[o] = m ? (y1 + t * (y2 - y1)) : 0.0f;
        vm[o] = m;
      }
    }

    // --- box1_in_box2(c1, c2): corners of box1 inside box2 ---
    {
      float ax = c2x[0], ay = c2y[0];
      float abx = c2x[1] - ax, aby = c2y[1] - ay;
      float adx = c2x[3] - ax, ady = c2y[3] - ay;
      float ab2 = abx * abx + aby * aby, ad2 = adx * adx + ady * ady;
#pragma unroll
      for (int k = 0; k < 4; ++k) {
        float apx = c1x[k] - ax, apy = c1y[k] - ay;
        float pab = (apx * abx + apy * aby) / ab2;
        float pad_ = (apx * adx + apy * ady) / ad2;
        vm[k] = (pab > -1e-6f) && (pab < 1.0f + 1e-6f) &&
                (pad_ > -1e-6f) && (pad_ < 1.0f + 1e-6f);
      }
    }
    // --- box1_in_box2(c2, c1): corners of box2 inside box1 ---
    {
      float ax = c1x[0], ay = c1y[0];
      float abx = c1x[1] - ax, aby = c1y[1] - ay;
      float adx = c1x[3] - ax, ady = c1y[3] - ay;
      float ab2 = abx * abx + aby * aby, ad2 = adx * adx + ady * ady;
#pragma unroll
      for (int k = 0; k < 4; ++k) {
        float apx = c2x[k] - ax, apy = c2y[k] - ay;
        float pab = (apx * abx + apy * aby) / ab2;
        float pad_ = (apx * adx + apy * ady) / ad2;
        vm[4 + k] = (pab > -1e-6f) && (pab < 1.0f + 1e-6f) &&
                    (pad_ > -1e-6f) && (pad_ < 1.0f + 1e-6f);
      }
    }

    // --- masked mean of valid vertices ---
    float sx = 0.0f, sy = 0.0f, cm = 0.0f;
#pragma unroll
    for (int k = 0; k < 24; ++k) {
      float mf = vm[k] ? 1.0f : 0.0f;
      sx += vx[k] * mf; sy += vy[k] * mf; cm += mf;
    }
    float nv = fmaxf(cm, 1.0f);
    float mx = sx / nv, my = sy / nv;

    // --- angles (invalid -> 1e9 so they sort to the end) ---
#pragma unroll
    for (int k = 0; k < 24; ++k)
      ang[k] = vm[k] ? atan2f(vy[k] - my, vx[k] - mx) : 1.0e9f;

    // --- stable insertion sort by angle (matches stable argsort) ---
    for (int a = 1; a < 24; ++a) {
      float ka = ang[a], xa = vx[a], ya = vy[a];
      bool ma = vm[a];
      int b = a - 1;
      while (b >= 0 && ang[b] > ka) {
        ang[b + 1] = ang[b]; vx[b + 1] = vx[b]; vy[b + 1] = vy[b]; vm[b + 1] = vm[b];
        --b;
      }
      ang[b + 1] = ka; vx[b + 1] = xa; vy[b + 1] = ya; vm[b + 1] = ma;
    }

    // --- replace invalid entries with the first sorted vertex ---
    float fx = vx[0], fy = vy[0];
#pragma unroll
    for (int k = 0; k < 24; ++k) {
      if (!vm[k]) { vx[k] = fx; vy[k] = fy; }
    }

    // --- shoelace area ---
    float s2 = 0.0f;
#pragma unroll
    for (int k = 0; k < 24; ++k) {
      int kn = (k == 23) ? 0 : (k + 1);
      s2 += vx[k] * vy[kn] - vx[kn] * vy[k];
    }
    float area = 0.5f * fabsf(s2);

    // --- 3D IoU + DIoU ---
    float zmax1 = pz + 0.5f * pd, zmin1 = pz - 0.5f * pd;
    float zmax2 = gz + 0.5f * gd, zmin2 = gz - 0.5f * gd;
    float zo = fmaxf(fminf(zmax1, zmax2) - fmaxf(zmin1, zmin2), 0.0f);
    float inter3 = area * zo;
    float vol1 = pw * ph * pd, vol2 = gw * gh * gd;
    float uni = vol1 + vol2 - inter3;
    float iou = inter3 / uni;
    float target = iou * 2.0f - 1.0f;

    float dx0 = px - gx, dy0 = py - gy, dz0 = pz - gz;
    float interdiag = dx0 * dx0 + dy0 * dy0 + dz0 * dz0;
    float outer_h = fmaxf(fmaxf(zmax1, zmax2) - fminf(zmin1, zmin2), 0.0f);
    float outx = fmaxf(fmaxf(c1x[2], c2x[2]) - fminf(c1x[0], c2x[0]), 0.0f);
    float outy = fmaxf(fmaxf(c1y[2], c2y[2]) - fminf(c1y[0], c2y[0]), 0.0f);
    float outerdiag = outx * outx + outy * outy + outer_h * outer_h;
    float diou = fminf(fmaxf(iou - interdiag / outerdiag, -1.0f), 1.0f);

    float mf = mask[idx] ? 1.0f : 0.0f;
    float predv = pred_iou[idx];
    contrib = (fabsf(predv - target) + (1.0f - diou)) * mf;
    cntf = mf;
  }

  // --- block reduction: wave32 shfl tree, then LDS across 8 waves ---
  float v = contrib, c = cntf;
#pragma unroll
  for (int off = 16; off > 0; off >>= 1) {
    v += __shfl_down(v, off, 32);
    c += __shfl_down(c, off, 32);
  }
  __shared__ float sN[8], sC[8];
  int lane = threadIdx.x & 31;
  int wid = threadIdx.x >> 5;
  if (lane == 0) { sN[wid] = v; sC[wid] = c; }
  __syncthreads();
  if (threadIdx.x == 0) {
    float a = 0.0f, b = 0.0f;
#pragma unroll
    for (int w = 0; w < 8; ++w) { a += sN[w]; b += sC[w]; }
    partials[blockIdx.x] = a;
    partials[nbp + blockIdx.x] = b;
  }
}

// ---------------------------------------------------------------------------
// Workspace zero-init (so partial-count padding to a multiple of 64 is zero,
// keeping the WMMA reduce loop divergence-free with EXEC = all ones).
// ---------------------------------------------------------------------------
__global__ void iou3d_loss_zero_kernel(float* __restrict__ ws, int n) {
  int i = blockIdx.x * blockDim.x + threadIdx.x;
  if (i < n) ws[i] = 0.0f;
}

// ---------------------------------------------------------------------------
// Final reduction on a single wave32 using V_WMMA_F32_16X16X4_F32.
// D = A(16x4 partials) x B(4x16 ones) + C accumulates 64 partials per WMMA:
// D[m,n] = sum_k A[m,k]. Per the CDNA5 f32 A layout, lane m supplies A[m,0]
// (VGPR0) and A[m,1] (VGPR1); lane m+16 supplies A[m,2], A[m,3]. After the
// loop, column 0 lives in acc[r]: lane0 = D[r,0], lane16 = D[8+r,0].
// ---------------------------------------------------------------------------
__global__ void iou3d_loss_reduce_kernel(const float* __restrict__ partials,
                                         int nbp, float* __restrict__ out) {
  int lane = threadIdx.x;           // 32 threads, one wave, no divergence
  v2f onesB = {1.0f, 1.0f};         // B = 4x16 all-ones
  v8f accN = {};                    // C/D accumulators (16x16 f32)
  v8f accC = {};

  int m = (lane < 16) ? lane : (lane - 16);
  int k0 = (lane < 16) ? 0 : 2;

  for (int base = 0; base < nbp; base += 64) {
    int i0 = base + m * 4 + k0;
    v2f a; a[0] = partials[i0]; a[1] = partials[i0 + 1];
    accN = __builtin_amdgcn_wmma_f32_16x16x4_f32(
        /*neg_a=*/false, a, /*neg_b=*/false, onesB,
        /*c_mod=*/(short)0, accN, /*reuse_a=*/false, /*reuse_b=*/false);
    v2f b; b[0] = partials[nbp + i0]; b[1] = partials[nbp + i0 + 1];
    accC = __builtin_amdgcn_wmma_f32_16x16x4_f32(
        false, b, false, onesB, (short)0, accC, false, false);
  }

  float sN = 0.0f, sC = 0.0f;
#pragma unroll
  for (int r = 0; r < 8; ++r) { sN += accN[r]; sC += accC[r]; }

  // column 0 of D: rows 0..7 in lane 0, rows 8..15 in lane 16
  float nTot = __shfl(sN, 0, 32) + __shfl(sN, 16, 32);
  float cTot = __shfl(sC, 0, 32) + __shfl(sC, 16, 32);
  if (lane == 0) out[0] = nTot / fmaxf(cTot, 1e-4f);
}

// ---------------------------------------------------------------------------
// Launch
// ---------------------------------------------------------------------------
extern "C" void kernel_launch(void* const* d_in, const int* in_sizes, int n_in,
                              void* d_out, int out_size, void* d_ws, size_t ws_size,
                              hipStream_t stream) {
  const float* pred_iou = (const float*)d_in[0];          // (B,N) f32
  const float* pred_boxes = (const float*)d_in[1];        // (B,N,7) f32
  const float* gt_boxes = (const float*)d_in[2];          // (B,N,8) f32
  const unsigned char* mask = (const unsigned char*)d_in[3]; // (B,N) bool
  float* out = (float*)d_out;
  float* partials = (float*)d_ws;

  int total = in_sizes[0];                 // B*N
  int blocks = (total + 255) / 256;        // 2048 for the reference shapes
  int nbp = (blocks + 63) & ~63;           // pad partial count to multiple of 64
  int zn = 2 * nbp;

  iou3d_loss_zero_kernel<<<(zn + 255) / 256, 256, 0, stream>>>(partials, zn);
  iou3d_loss_main_kernel<<<blocks, 256, 0, stream>>>(
      pred_iou, pred_boxes, gt_boxes, mask, partials, total, nbp);
  iou3d_loss_reduce_kernel<<<1, 32, 0, stream>>>(partials, nbp, out);
}